// ProposalLayer_20512763806374
// MI455X (gfx1250) — compile-verified
//
#include <hip/hip_runtime.h>

// ProposalLayer for MI455X (gfx1250, wave32).
// Stages: histogram top-k selection -> LDS bitonic sort -> box decode ->
// WMMA-accelerated pairwise IoU suppression-mask build -> sequential greedy NMS.
// V_WMMA_F32_16X16X4_F32 computes the 16x16 tile of (area_i + area_j) as a
// rank-2 outer product: A=[area_row, 1, 0, 0], B=[[1..1],[area_col],0,0].

#define NBATCH   2
#define KTOP     6000           // PRE_NMS_LIMIT (375 * 16 exactly)
#define PROP     2000           // PROPOSAL_COUNT
#define WORDS    94             // ceil(KTOP/64)
#define CAP      8192           // candidate capacity (64KB LDS sort)
#define NMS_T    0.7f
#define SEL_K    6000

typedef __attribute__((ext_vector_type(2))) float v2f;
typedef __attribute__((ext_vector_type(8))) float v8f;

__device__ __forceinline__ unsigned sortable_key(float f) {
  unsigned u = __float_as_uint(f);
  return (u & 0x80000000u) ? ~u : (u | 0x80000000u);
}

// ---------------- zero workspace + output ----------------
__global__ void k_zero(unsigned* hist, int histWords, int* count,
                       unsigned long long* masks, long long maskWords,
                       float* out, int outWords) {
  long long stride = (long long)gridDim.x * blockDim.x;
  long long i0 = (long long)blockIdx.x * blockDim.x + threadIdx.x;
  for (long long i = i0; i < histWords; i += stride) hist[i] = 0u;
  for (long long i = i0; i < NBATCH; i += stride) count[i] = 0;
  for (long long i = i0; i < maskWords; i += stride) masks[i] = 0ULL;
  for (long long i = i0; i < outWords; i += stride) out[i] = 0.0f;
}

// ---------------- 64K-bin histogram of top 16 key bits ----------------
__global__ void k_hist(const float* __restrict__ probs, unsigned* __restrict__ hist, int N) {
  long long gid = (long long)blockIdx.x * blockDim.x + threadIdx.x;
  if (gid >= (long long)NBATCH * N) return;
  int b = (int)(gid / N);
  float s = probs[(size_t)gid * 2 + 1];
  unsigned key = sortable_key(s);
  atomicAdd(&hist[(size_t)b * 65536 + (key >> 16)], 1u);
}

// ---------------- find per-batch threshold bucket ----------------
__global__ __launch_bounds__(1024) void k_thresh(const unsigned* __restrict__ hist,
                                                 unsigned* __restrict__ thresh) {
  __shared__ unsigned psum[1024];
  int b = blockIdx.x;
  int tid = threadIdx.x;
  const unsigned* h = hist + (size_t)b * 65536;
  unsigned mySum = 0;
  int base = tid * 64;                     // reversed (descending-bucket) order
  for (int i = 0; i < 64; ++i) mySum += h[65535 - (base + i)];
  psum[tid] = mySum;
  __syncthreads();
  if (tid == 0) {                          // exclusive scan (tiny, serial)
    unsigned run = 0;
    for (int t = 0; t < 1024; ++t) { unsigned v = psum[t]; psum[t] = run; run += v; }
  }
  __syncthreads();
  unsigned prefix = psum[tid];
  if (prefix < SEL_K && prefix + mySum >= SEL_K) {
    unsigned run = prefix;
    for (int i = 0; i < 64; ++i) {
      run += h[65535 - (base + i)];
      if (run >= SEL_K) { thresh[b] = (unsigned)(65535 - (base + i)); break; }
    }
  }
}

// ---------------- compact survivors: key<<32 | (~index) ----------------
__global__ void k_compact(const float* __restrict__ probs, const unsigned* __restrict__ thresh,
                          int* __restrict__ count, unsigned long long* __restrict__ cand, int N) {
  long long gid = (long long)blockIdx.x * blockDim.x + threadIdx.x;
  if (gid >= (long long)NBATCH * N) return;
  int b = (int)(gid / N);
  unsigned n = (unsigned)(gid % N);
  unsigned key = sortable_key(probs[(size_t)gid * 2 + 1]);
  if ((key >> 16) >= thresh[b]) {
    int pos = atomicAdd(&count[b], 1);
    if (pos < CAP)
      cand[(size_t)b * CAP + pos] =
          ((unsigned long long)key << 32) | (unsigned long long)(0xFFFFFFFFu - n);
  }
}

// ---------------- per-batch bitonic sort (desc) + box decode ----------------
__global__ __launch_bounds__(1024) void k_sort_decode(const unsigned long long* __restrict__ cand,
                                                      const int* __restrict__ count,
                                                      const float* __restrict__ bbox,
                                                      const float* __restrict__ anch,
                                                      float* __restrict__ boxes, int N) {
  __shared__ unsigned long long sd[CAP];   // 64KB
  int b = blockIdx.x;
  int tid = threadIdx.x;
  int cnt = count[b]; if (cnt > CAP) cnt = CAP;
  for (int i = tid; i < CAP; i += 1024)
    sd[i] = (i < cnt) ? cand[(size_t)b * CAP + i] : 0ULL;  // 0 sorts to the end
  __syncthreads();
  for (int k = 2; k <= CAP; k <<= 1) {
    for (int j = k >> 1; j > 0; j >>= 1) {
      for (int i = tid; i < CAP; i += 1024) {
        int ixj = i ^ j;
        if (ixj > i) {
          unsigned long long a = sd[i], c = sd[ixj];
          bool sw = ((i & k) == 0) ? (a < c) : (a > c);   // overall descending
          if (sw) { sd[i] = c; sd[ixj] = a; }
        }
      }
      __syncthreads();
    }
  }
  // decode top KTOP boxes (score order)
  for (int i = tid; i < KTOP; i += 1024) {
    unsigned n = 0xFFFFFFFFu - (unsigned)(sd[i] & 0xFFFFFFFFULL);
    size_t base = ((size_t)b * N + n) * 4;
    float d0 = bbox[base + 0] * 0.1f, d1 = bbox[base + 1] * 0.1f;
    float d2 = bbox[base + 2] * 0.2f, d3 = bbox[base + 3] * 0.2f;
    float a0 = anch[base + 0], a1 = anch[base + 1];
    float a2 = anch[base + 2], a3 = anch[base + 3];
    float h = a2 - a0, w = a3 - a1;
    float cy = a0 + 0.5f * h + d0 * h;
    float cx = a1 + 0.5f * w + d1 * w;
    h *= expf(d2);
    w *= expf(d3);
    float y1 = cy - 0.5f * h, x1 = cx - 0.5f * w;
    float y2 = y1 + h, x2 = x1 + w;
    size_t ob = ((size_t)b * KTOP + i) * 4;
    boxes[ob + 0] = fminf(fmaxf(y1, 0.f), 1.f);
    boxes[ob + 1] = fminf(fmaxf(x1, 0.f), 1.f);
    boxes[ob + 2] = fminf(fmaxf(y2, 0.f), 1.f);
    boxes[ob + 3] = fminf(fmaxf(x2, 0.f), 1.f);
  }
}

// ---------------- WMMA IoU suppression-mask build ----------------
// grid (KTOP/16, WORDS, B), 128 threads = 4 wave32; wave w -> 16x16 tile
// rows [16*bx, +16), cols [64*by + 16*w, +16).
__global__ __launch_bounds__(128) void k_mask(const float* __restrict__ boxes,
                                              unsigned long long* __restrict__ masks) {
  int b = blockIdx.z;
  int rowStart = blockIdx.x * 16;
  int colBase  = blockIdx.y * 64;
  if (colBase + 64 <= rowStart) return;    // only j > i can suppress
  __shared__ float rb[16][4];
  __shared__ float cb[64][4];
  __shared__ float rArea[16];
  __shared__ float cArea[64];
  const float* Bx = boxes + (size_t)b * KTOP * 4;
  int tid = threadIdx.x;
  if (tid < 64) { int r = tid >> 2, c = tid & 3; rb[r][c] = Bx[(size_t)(rowStart + r) * 4 + c]; }
  for (int t = tid; t < 256; t += 128) {
    int r = t >> 2, c = t & 3; int gc = colBase + r;
    cb[r][c] = (gc < KTOP) ? Bx[(size_t)gc * 4 + c] : 0.0f;
  }
  __syncthreads();
  if (tid < 16) rArea[tid] = (rb[tid][2] - rb[tid][0]) * (rb[tid][3] - rb[tid][1]);
  if (tid < 64) cArea[tid] = (cb[tid][2] - cb[tid][0]) * (cb[tid][3] - cb[tid][1]);
  __syncthreads();

  int lane = tid & 31;
  int wave = tid >> 5;
  // A (16x4): col0 = rowArea, col1 = 1   |   B (4x16): row0 = 1, row1 = colArea
  v2f av, bv;
  av.x = (lane < 16) ? rArea[lane] : 0.0f;            // K=0 (lanes<16) / K=2 -> 0
  av.y = (lane < 16) ? 1.0f : 0.0f;                   // K=1 / K=3 -> 0
  bv.x = (lane < 16) ? 1.0f : 0.0f;                   // B[0,n]=1
  bv.y = (lane < 16) ? cArea[wave * 16 + lane] : 0.0f;// B[1,n]=colArea
  v8f cz = {};
  v8f dsum = __builtin_amdgcn_wmma_f32_16x16x4_f32(false, av, false, bv,
                                                   (short)0, cz, false, false);
  int colIn = lane & 15;
  int gcol  = colBase + wave * 16 + colIn;
  float cy1 = cb[wave * 16 + colIn][0], cx1 = cb[wave * 16 + colIn][1];
  float cy2 = cb[wave * 16 + colIn][2], cx2 = cb[wave * 16 + colIn][3];
  int rOff = (lane >= 16) ? 8 : 0;
  unsigned long long* mb = masks + (size_t)b * KTOP * WORDS;
  int word  = blockIdx.y;
  int shift = wave * 16;
  for (int r = 0; r < 8; ++r) {
    int grow = rowStart + r + rOff;
    float ry1 = rb[r + rOff][0], rx1 = rb[r + rOff][1];
    float ry2 = rb[r + rOff][2], rx2 = rb[r + rOff][3];
    float iy = fminf(ry2, cy2) - fmaxf(ry1, cy1);
    float ix = fminf(rx2, cx2) - fmaxf(rx1, cx1);
    float inter = fmaxf(iy, 0.0f) * fmaxf(ix, 0.0f);
    float uni = dsum[r] - inter;                      // area_i + area_j - inter (WMMA)
    float iou = (uni > 0.0f) ? inter / uni : 0.0f;
    bool pred = (iou > NMS_T) && (gcol > grow) && (gcol < KTOP);
    unsigned long long bal = __ballot((int)pred);     // wave32: bits [31:0]
    if (lane == 0) {
      unsigned long long lo = bal & 0xFFFFull;          // row rowStart+r
      unsigned long long hi = (bal >> 16) & 0xFFFFull;  // row rowStart+r+8
      if (lo) atomicOr(&mb[(size_t)(rowStart + r) * WORDS + word], lo << shift);
      if (hi) atomicOr(&mb[(size_t)(rowStart + r + 8) * WORDS + word], hi << shift);
    }
  }
}

// ---------------- sequential greedy NMS reduce ----------------
__global__ __launch_bounds__(128) void k_reduce(const float* __restrict__ boxes,
                                                const unsigned long long* __restrict__ masks,
                                                float* __restrict__ out) {
  __shared__ unsigned long long removed[WORDS];
  __shared__ int keepCount;
  __shared__ int curKeep;
  int b = blockIdx.x;
  int tid = threadIdx.x;
  for (int i = tid; i < WORDS; i += blockDim.x) removed[i] = 0ULL;
  if (tid == 0) keepCount = 0;
  __syncthreads();
  const float* Bx = boxes + (size_t)b * KTOP * 4;
  const unsigned long long* Mb = masks + (size_t)b * KTOP * WORDS;
  float* Ob = out + (size_t)b * PROP * 4;
  for (int i = 0; i < KTOP; ++i) {
    if (tid == 0) {
      bool rem = (removed[i >> 6] >> (i & 63)) & 1ULL;
      int k = -1;
      if (!rem) k = keepCount++;
      curKeep = k;
      if (k >= 0 && k < PROP) {
        Ob[(size_t)k * 4 + 0] = Bx[(size_t)i * 4 + 0];
        Ob[(size_t)k * 4 + 1] = Bx[(size_t)i * 4 + 1];
        Ob[(size_t)k * 4 + 2] = Bx[(size_t)i * 4 + 2];
        Ob[(size_t)k * 4 + 3] = Bx[(size_t)i * 4 + 3];
      }
    }
    __syncthreads();
    if (curKeep >= 0) {
      for (int w = tid; w < WORDS; w += blockDim.x)
        removed[w] |= Mb[(size_t)i * WORDS + w];
    }
    __syncthreads();
    if (keepCount >= PROP) break;   // remaining iterations cannot change output
  }
}

extern "C" void kernel_launch(void* const* d_in, const int* in_sizes, int n_in,
                              void* d_out, int out_size, void* d_ws, size_t ws_size,
                              hipStream_t stream) {
  const float* probs = (const float*)d_in[0];
  const float* bbox  = (const float*)d_in[1];
  const float* anch  = (const float*)d_in[2];
  float* out = (float*)d_out;
  const int N = in_sizes[0] / (NBATCH * 2);   // 262144

  char* ws = (char*)d_ws;
  size_t off = 0;
  unsigned* hist = (unsigned*)(ws + off);              off += (size_t)NBATCH * 65536 * 4;
  int* count = (int*)(ws + off);                       off += 16;
  unsigned* thresh = (unsigned*)(ws + off);            off += 16;
  unsigned long long* cand = (unsigned long long*)(ws + off); off += (size_t)NBATCH * CAP * 8;
  float* boxes = (float*)(ws + off);                   off += (size_t)NBATCH * KTOP * 4 * 4;
  unsigned long long* masks = (unsigned long long*)(ws + off);
  // total ws use ~10 MB

  long long maskWords = (long long)NBATCH * KTOP * WORDS;
  k_zero<<<4096, 256, 0, stream>>>(hist, NBATCH * 65536, count, masks, maskWords,
                                   out, NBATCH * PROP * 4);

  int total = NBATCH * N;
  int blocks = (total + 255) / 256;
  k_hist<<<blocks, 256, 0, stream>>>(probs, hist, N);
  k_thresh<<<NBATCH, 1024, 0, stream>>>(hist, thresh);
  k_compact<<<blocks, 256, 0, stream>>>(probs, thresh, count, cand, N);
  k_sort_decode<<<NBATCH, 1024, 0, stream>>>(cand, count, bbox, anch, boxes, N);
  k_mask<<<dim3(KTOP / 16, WORDS, NBATCH), 128, 0, stream>>>(boxes, masks);
  k_reduce<<<NBATCH, 128, 0, stream>>>(boxes, masks, out);
}